// GraphIsomorphismLayer_27315992003076
// MI455X (gfx1250) — compile-verified
//
#include <hip/hip_runtime.h>
#include <hip/hip_bf16.h>

typedef __attribute__((ext_vector_type(2))) float v2f;
typedef __attribute__((ext_vector_type(8))) float v8f;

#define DIM   256
#define H1    64
#define H2    32

// ---------------------------------------------------------------------------
// Kernel 1: y = w[0] * x   (initialize accumulation buffer with self-loop term)
// ---------------------------------------------------------------------------
__global__ void gin_init_kernel(const float* __restrict__ x,
                                const float* __restrict__ w,
                                float* __restrict__ y,
                                long long total4) {
    long long id = (long long)blockIdx.x * blockDim.x + threadIdx.x;
    if (id >= total4) return;
    float s = w[0];
    const float4* xv = (const float4*)x;
    float4* yv = (float4*)y;
    float4 v = xv[id];
    v.x *= s; v.y *= s; v.z *= s; v.w *= s;
    yv[id] = v;
}

// ---------------------------------------------------------------------------
// Kernel 2: edge scatter-add: y[src] += x[dst]
// One wave (32 lanes) per edge; each lane owns 8 consecutive floats (32B),
// so the gather of x[dst] is one fully-coalesced 1KB row per wave.
// x (100MB) and y (100MB) both fit in the 192MB L2, so gathers + f32 atomics
// are serviced from L2.
// ---------------------------------------------------------------------------
__global__ void gin_scatter_kernel(const float* __restrict__ x,
                                   const int* __restrict__ esrc,
                                   const int* __restrict__ edst,
                                   float* __restrict__ y,
                                   long long n_items) {
    long long id = (long long)blockIdx.x * blockDim.x + threadIdx.x;
    if (id >= n_items) return;
    long long edge  = id >> 5;          // 32 lanes per edge
    int       chunk = (int)(id & 31) * 8;
    int s = esrc[edge];
    int d = edst[edge];
    const float* src = x + (size_t)d * DIM + chunk;
    float*       dst = y + (size_t)s * DIM + chunk;
    float4 v0 = *(const float4*)(src + 0);
    float4 v1 = *(const float4*)(src + 4);
    __hip_atomic_fetch_add(dst + 0, v0.x, __ATOMIC_RELAXED, __HIP_MEMORY_SCOPE_AGENT);
    __hip_atomic_fetch_add(dst + 1, v0.y, __ATOMIC_RELAXED, __HIP_MEMORY_SCOPE_AGENT);
    __hip_atomic_fetch_add(dst + 2, v0.z, __ATOMIC_RELAXED, __HIP_MEMORY_SCOPE_AGENT);
    __hip_atomic_fetch_add(dst + 3, v0.w, __ATOMIC_RELAXED, __HIP_MEMORY_SCOPE_AGENT);
    __hip_atomic_fetch_add(dst + 4, v1.x, __ATOMIC_RELAXED, __HIP_MEMORY_SCOPE_AGENT);
    __hip_atomic_fetch_add(dst + 5, v1.y, __ATOMIC_RELAXED, __HIP_MEMORY_SCOPE_AGENT);
    __hip_atomic_fetch_add(dst + 6, v1.z, __ATOMIC_RELAXED, __HIP_MEMORY_SCOPE_AGENT);
    __hip_atomic_fetch_add(dst + 7, v1.w, __ATOMIC_RELAXED, __HIP_MEMORY_SCOPE_AGENT);
}

// ---------------------------------------------------------------------------
// Kernel 3: fused MLP readout with WMMA f32 16x16x4 + residual.
// One wave (32 lanes) owns one 16-node tile. 4 waves per block.
// Multi-tile accumulation: one A load feeds 4 (or 2) WMMAs, so each
// activation element is loaded once per layer (once per group of 4 tiles in layer 3).
//
// VGPR layouts per CDNA5 ISA (7.12.2), wave32:
//   A (16x4 f32, 2 VGPRs): lane L -> m = L&15, kh = L>>4;
//       a[0] = A[m][k0 + 2*kh], a[1] = A[m][k0 + 2*kh + 1]
//   B (4x16 f32, 2 VGPRs): lane L -> n = L&15, kh = L>>4;
//       b[j] = B[k0 + 2*kh + j][n] = W[n][k0 + 2*kh + j]   (B = W^T)
//   C/D (16x16 f32, 8 VGPRs): acc[j] = D[j + 8*kh][n]
// ---------------------------------------------------------------------------
__global__ __launch_bounds__(128) void gin_mlp_kernel(
        const float* __restrict__ y,
        const float* __restrict__ x,
        const float* __restrict__ W1,   // [64][256]
        const float* __restrict__ W2,   // [32][64]
        const float* __restrict__ W3,   // [256][32]
        float* __restrict__ out,
        int n_nodes) {
    __shared__ __align__(16) float lds_h1[4][16][H1];
    __shared__ __align__(16) float lds_h2[4][16][H2];

    const int wave = threadIdx.x >> 5;
    const int lane = threadIdx.x & 31;
    const int tile = blockIdx.x * 4 + wave;
    const int row0 = tile * 16;
    if (row0 >= n_nodes) return;     // whole wave exits: EXEC stays all-ones

    const int nl = lane & 15;        // m (for A) / n (for B,D) within 16
    const int kh = lane >> 4;        // K-half selector

    // ---- Layer 1: [16 x 256] @ W1^T -> relu -> lds_h1 [16 x 64] ----
    {
        const float* arow = y + (size_t)(row0 + nl) * DIM + 2 * kh;
        const float* wbase = W1 + (size_t)nl * DIM + 2 * kh;
        v8f acc[4];
#pragma unroll
        for (int t = 0; t < 4; ++t) acc[t] = (v8f){};
        for (int k0 = 0; k0 < DIM; k0 += 4) {
            v2f a = *(const v2f*)(arow + k0);
#pragma unroll
            for (int t = 0; t < 4; ++t) {
                v2f b = *(const v2f*)(wbase + (size_t)(t * 16) * DIM + k0);
                acc[t] = __builtin_amdgcn_wmma_f32_16x16x4_f32(
                        false, a, false, b, (short)0, acc[t], false, false);
            }
        }
#pragma unroll
        for (int t = 0; t < 4; ++t)
#pragma unroll
            for (int j = 0; j < 8; ++j) {
                float v = acc[t][j];
                lds_h1[wave][j + 8 * kh][t * 16 + nl] = v > 0.f ? v : 0.f;
            }
    }

    // ---- Layer 2: [16 x 64] @ W2^T -> relu -> lds_h2 [16 x 32] ----
    {
        const float* arow = &lds_h1[wave][nl][2 * kh];
        const float* wbase = W2 + (size_t)nl * H1 + 2 * kh;
        v8f acc[2];
#pragma unroll
        for (int t = 0; t < 2; ++t) acc[t] = (v8f){};
        for (int k0 = 0; k0 < H1; k0 += 4) {
            v2f a = *(const v2f*)(arow + k0);
#pragma unroll
            for (int t = 0; t < 2; ++t) {
                v2f b = *(const v2f*)(wbase + (size_t)(t * 16) * H1 + k0);
                acc[t] = __builtin_amdgcn_wmma_f32_16x16x4_f32(
                        false, a, false, b, (short)0, acc[t], false, false);
            }
        }
#pragma unroll
        for (int t = 0; t < 2; ++t)
#pragma unroll
            for (int j = 0; j < 8; ++j) {
                float v = acc[t][j];
                lds_h2[wave][j + 8 * kh][t * 16 + nl] = v > 0.f ? v : 0.f;
            }
    }

    // ---- Layer 3: [16 x 32] @ W3^T -> relu -> + x residual -> out [16 x 256]
    // 16 output tiles processed in 4 groups of 4 (32 accumulator VGPRs/group).
    {
        const float* arow = &lds_h2[wave][nl][2 * kh];
        for (int g = 0; g < 4; ++g) {
            const float* wbase = W3 + (size_t)(g * 64 + nl) * H2 + 2 * kh;
            v8f acc[4];
#pragma unroll
            for (int t = 0; t < 4; ++t) acc[t] = (v8f){};
            for (int k0 = 0; k0 < H2; k0 += 4) {
                v2f a = *(const v2f*)(arow + k0);
#pragma unroll
                for (int t = 0; t < 4; ++t) {
                    v2f b = *(const v2f*)(wbase + (size_t)(t * 16) * H2 + k0);
                    acc[t] = __builtin_amdgcn_wmma_f32_16x16x4_f32(
                            false, a, false, b, (short)0, acc[t], false, false);
                }
            }
#pragma unroll
            for (int t = 0; t < 4; ++t)
#pragma unroll
                for (int j = 0; j < 8; ++j) {
                    int m   = j + 8 * kh;
                    int col = (g * 4 + t) * 16 + nl;
                    size_t idx = (size_t)(row0 + m) * DIM + col;
                    float v = acc[t][j];
                    v = v > 0.f ? v : 0.f;
                    out[idx] = v + x[idx];
                }
        }
    }
}

// ---------------------------------------------------------------------------
extern "C" void kernel_launch(void* const* d_in, const int* in_sizes, int n_in,
                              void* d_out, int out_size, void* d_ws, size_t ws_size,
                              hipStream_t stream) {
    const float* x    = (const float*)d_in[0];
    const int*   esrc = (const int*)  d_in[1];
    const int*   edst = (const int*)  d_in[2];
    const float* w    = (const float*)d_in[3];
    const float* W1   = (const float*)d_in[4];
    const float* W2   = (const float*)d_in[5];
    const float* W3   = (const float*)d_in[6];
    float*       out  = (float*)d_out;
    float*       y    = (float*)d_ws;   // [n_nodes][256] accumulation buffer

    const int n_nodes = in_sizes[0] / DIM;
    const int n_edges = in_sizes[1];

    // 1) y = w * x
    {
        long long total4 = (long long)n_nodes * (DIM / 4);
        int blocks = (int)((total4 + 255) / 256);
        gin_init_kernel<<<blocks, 256, 0, stream>>>(x, w, y, total4);
    }
    // 2) y[src] += x[dst] over all edges (one wave per edge)
    {
        long long n_items = (long long)n_edges * 32;
        long long blocks  = (n_items + 255) / 256;
        gin_scatter_kernel<<<(int)blocks, 256, 0, stream>>>(x, esrc, edst, y, n_items);
    }
    // 3) fused MLP + residual
    {
        int tiles  = (n_nodes + 15) / 16;
        int blocks = (tiles + 3) / 4;
        gin_mlp_kernel<<<blocks, 128, 0, stream>>>(y, x, W1, W2, W3, out, n_nodes);
    }
}